// CC_DC_and_CE_loss_32134945308794
// MI455X (gfx1250) — compile-verified
//
#include <hip/hip_runtime.h>

#define NSEG   16
#define REC    96
#define SMOOTHF 1e-5f

typedef __attribute__((ext_vector_type(2))) float v2f;
typedef __attribute__((ext_vector_type(8))) float v8f;

__device__ __forceinline__ float waveRedSum(float v) {
#pragma unroll
  for (int off = 16; off > 0; off >>= 1)
    v += __shfl_xor(v, off, 32);
  return v;
}

// ---------------------------------------------------------------------------
// Pass 1: stream 118 MB once; per-block 96-float partial record -> d_ws.
// Blocks 0..G-1 cover batch 0, G..2G-1 cover batch 1, each a contiguous chunk.
// ---------------------------------------------------------------------------
__global__ __launch_bounds__(256)
void cc_loss_pass1(const float* __restrict__ logits,
                   const int*   __restrict__ target,
                   const int*   __restrict__ lbl,
                   const int*   __restrict__ vor,
                   float*       __restrict__ rec,
                   int G, int iters)
{
  __shared__ float segacc[NSEG * 5]; // per segment: tpc, cntl, fpc, cec, cnt
  __shared__ float scal[13];         // ce, sp[4], tp[4], ct[4]
  const int tid = threadIdx.x;
  if (tid < NSEG * 5) segacc[tid] = 0.f;
  if (tid < 13)       scal[tid]   = 0.f;
  __syncthreads();

  const long long V     = 128ll * 128 * 128;
  const int       batch = blockIdx.x / G;
  const int       g     = blockIdx.x - batch * G;
  const long long vbase = (long long)g * iters * 256;

  const float* __restrict__ c0p = logits + (long long)(batch * 4 + 0) * V;
  const float* __restrict__ c1p = logits + (long long)(batch * 4 + 1) * V;
  const float* __restrict__ c2p = logits + (long long)(batch * 4 + 2) * V;
  const float* __restrict__ c3p = logits + (long long)(batch * 4 + 3) * V;
  const int*   __restrict__ tg  = target + batch * V;
  const int*   __restrict__ lb  = lbl    + batch * V;
  const int*   __restrict__ vo  = vor    + batch * V;

  float ce_sum = 0.f;
  float sp[4] = {0.f, 0.f, 0.f, 0.f};
  float tp[4] = {0.f, 0.f, 0.f, 0.f};
  float ct[4] = {0.f, 0.f, 0.f, 0.f};

  for (int k = 0; k < iters; ++k) {
    long long v = vbase + (long long)k * 256 + tid;

    if (k + 4 < iters) { // gfx1250 global_prefetch_b8 on the 4 logit streams
      long long vp = v + 4ll * 256;
      __builtin_prefetch(c0p + vp, 0, 3);
      __builtin_prefetch(c1p + vp, 0, 3);
      __builtin_prefetch(c2p + vp, 0, 3);
      __builtin_prefetch(c3p + vp, 0, 3);
    }

    float a0 = c0p[v], a1 = c1p[v], a2 = c2p[v], a3 = c3p[v];
    int   t  = tg[v];
    int   L  = lb[v];
    int   S  = vo[v];

    float m  = fmaxf(fmaxf(a0, a1), fmaxf(a2, a3));
    float e0 = __expf(a0 - m), e1 = __expf(a1 - m);
    float e2 = __expf(a2 - m), e3 = __expf(a3 - m);
    float ssum = e0 + e1 + e2 + e3;
    float inv  = 1.0f / ssum;
    float q0 = e0 * inv, q1 = e1 * inv, q2 = e2 * inv, q3 = e3 * inv;
    float lse = __logf(ssum);
    float lt  = (t == 0) ? a0 : (t == 1) ? a1 : (t == 2) ? a2 : a3;
    float ce  = (m + lse) - lt; // -log p[target]

    ce_sum += ce;
    sp[0] += q0; sp[1] += q1; sp[2] += q2; sp[3] += q3;
    if      (t == 0) { tp[0] += q0; ct[0] += 1.f; }
    else if (t == 1) { tp[1] += q1; ct[1] += 1.f; }
    else if (t == 2) { tp[2] += q2; ct[2] += 1.f; }
    else             { tp[3] += q3; ct[3] += 1.f; }

    // ---- vor-indexed sums: fpc (p1 where lbl==0), cec (ce), cnt ----
    int Sf = __builtin_amdgcn_readfirstlane(S);
    if (__all(S == Sf)) { // uniform fast path (true for this data layout)
      if (Sf >= 1 && Sf <= NSEG) {
        float rfp = waveRedSum((L == 0) ? q1 : 0.f);
        float rce = waveRedSum(ce);
        if ((tid & 31) == 0) {
          float* sa = &segacc[(Sf - 1) * 5];
          atomicAdd(sa + 2, rfp);
          atomicAdd(sa + 3, rce);
          atomicAdd(sa + 4, 32.f);
        }
      }
    } else {
      if (S >= 1 && S <= NSEG) {
        float* sa = &segacc[(S - 1) * 5];
        atomicAdd(sa + 2, (L == 0) ? q1 : 0.f);
        atomicAdd(sa + 3, ce);
        atomicAdd(sa + 4, 1.f);
      }
    }

    // ---- lbl-indexed sums: tpc (p1), cntl (count), lbl in {0} U {seg} ----
    unsigned long long nz = __ballot(L != 0);
    if (nz) {
      int fl = (int)__ffsll(nz) - 1;
      int Lf = __shfl(L, fl, 32);
      if (__all(L == 0 || L == Lf)) {
        if (Lf >= 1 && Lf <= NSEG) {
          float rtp = waveRedSum((L != 0) ? q1 : 0.f);
          float rcn = waveRedSum((L != 0) ? 1.f : 0.f);
          if ((tid & 31) == 0) {
            float* sa = &segacc[(Lf - 1) * 5];
            atomicAdd(sa + 0, rtp);
            atomicAdd(sa + 1, rcn);
          }
        }
      } else {
        if (L >= 1 && L <= NSEG) {
          float* sa = &segacc[(L - 1) * 5];
          atomicAdd(sa + 0, q1);
          atomicAdd(sa + 1, 1.f);
        }
      }
    }
  }

  // block-level reduction of the 13 per-thread scalars
  float vals[13] = { ce_sum, sp[0], sp[1], sp[2], sp[3],
                     tp[0], tp[1], tp[2], tp[3],
                     ct[0], ct[1], ct[2], ct[3] };
#pragma unroll
  for (int j = 0; j < 13; ++j) {
    float r = waveRedSum(vals[j]);
    if ((tid & 31) == 0) atomicAdd(&scal[j], r);
  }
  __syncthreads();

  float* rp = rec + (long long)blockIdx.x * REC;
  if      (tid < 13)  rp[tid] = scal[tid];
  else if (tid < 16)  rp[tid] = 0.f;              // padding
  else if (tid < REC) rp[tid] = segacc[tid - 16]; // 16 segs x 5 fields
}

// ---------------------------------------------------------------------------
// Pass 2: column-sum the (G x 96) partial matrix per batch with
// V_WMMA_F32_16X16X4_F32 (A = ones(16x4) -> D rows = exact f32 column sums;
// summing over all K makes the K lane-layout irrelevant). Then thread 0
// does the scalar Dice/CE combine.
// ---------------------------------------------------------------------------
__global__ __launch_bounds__(384)
void cc_loss_pass2(const float* __restrict__ rec,
                   float* __restrict__ out, int G)
{
  __shared__ float colsum[2][REC];
  const int tid  = threadIdx.x;
  const int wave = tid >> 5;
  const int lane = tid & 31;

  if (wave < 12) {
    const int batch = wave / 6;
    const int c0    = (wave % 6) * 16;
    const int half  = lane >> 4;
    const int n     = lane & 15;
    const float* base = rec + (long long)batch * G * REC;

    v8f acc = {};
    v2f a;  a.x = 1.0f; a.y = 1.0f; // ones A-matrix
    for (int g2 = 0; g2 < G; g2 += 4) {
      v2f b;
      b.x = base[(long long)(g2 + half)     * REC + c0 + n];
      b.y = base[(long long)(g2 + 2 + half) * REC + c0 + n];
      acc = __builtin_amdgcn_wmma_f32_16x16x4_f32(
          /*neg_a=*/false, a, /*neg_b=*/false, b,
          /*c_mod=*/(short)0, acc, /*reuse_a=*/false, /*reuse_b=*/false);
    }
    if (lane < 16) colsum[batch][c0 + lane] = acc[0]; // VGPR0, M=0 row
  }
  __syncthreads();

  if (tid == 0) {
    const float V = 128.f * 128.f * 128.f;
    float ce_tot = 0.f, dice_sum = 0.f, cc_sum = 0.f;
    for (int b = 0; b < 2; ++b) {
      const float* r = colsum[b];
      ce_tot += r[0];
      for (int c = 1; c < 4; ++c) {
        float tpv = r[5 + c], spv = r[1 + c], ctv = r[9 + c];
        float fpv = spv - tpv, fnv = ctv - tpv;
        float dc = (2.f * tpv + SMOOTHF) /
                   fmaxf(2.f * tpv + fpv + fnv + SMOOTHF, 1e-8f);
        dice_sum += dc;
      }
      for (int s = 0; s < NSEG; ++s) {
        float tpc  = r[16 + s * 5 + 0];
        float cntl = r[16 + s * 5 + 1];
        float fpc  = r[16 + s * 5 + 2];
        float cec  = r[16 + s * 5 + 3];
        float cnt  = r[16 + s * 5 + 4];
        float fnc  = cntl - tpc;
        float dcc  = (2.f * tpc + SMOOTHF) /
                     fmaxf(2.f * tpc + fnc + fpc + SMOOTHF, 1e-8f);
        float cet  = cec / fmaxf(cnt, 1.f);
        cc_sum += (cet - dcc);
      }
    }
    float ce_global   = ce_tot / (2.f * V);
    float dice_global = -(dice_sum / 6.f);   // mean over B=2, classes 1..3
    float cc_term     = cc_sum / 32.f;       // mean over B=2, 16 segments
    out[0] = ce_global + dice_global + cc_term;
  }
}

// ---------------------------------------------------------------------------
extern "C" void kernel_launch(void* const* d_in, const int* in_sizes, int n_in,
                              void* d_out, int out_size, void* d_ws, size_t ws_size,
                              hipStream_t stream)
{
  (void)in_sizes; (void)n_in; (void)out_size;
  const float* logits = (const float*)d_in[0];
  const int*   target = (const int*)d_in[1];
  const int*   lbl    = (const int*)d_in[2];
  const int*   vor    = (const int*)d_in[3];
  // d_in[4] = n_cc (==16, hardcoded as NSEG)

  const long long V = 128ll * 128 * 128;
  int G = 1024; // blocks per batch (power of two, multiple of 4)
  while (G > 4 && (size_t)(2ll * G * REC * sizeof(float)) > ws_size) G >>= 1;
  int iters = (int)(V / ((long long)G * 256));

  float* rec = (float*)d_ws; // 2*G*REC floats of partials

  hipLaunchKernelGGL(cc_loss_pass1, dim3(2 * G), dim3(256), 0, stream,
                     logits, target, lbl, vor, rec, G, iters);
  hipLaunchKernelGGL(cc_loss_pass2, dim3(1), dim3(384), 0, stream,
                     rec, (float*)d_out, G);
}